// YOLOLoss_53592601920096
// MI455X (gfx1250) — compile-verified
//
#include <hip/hip_runtime.h>
#include <math.h>

#define B_   16
#define A_   3
#define C_   80
#define H_   76
#define W_   76
#define T_   50
#define HW_  (H_ * W_)
#define N_   (B_ * A_ * HW_)   // 277248
#define NT_  (B_ * T_)         // 800
#define NBLK 64
#define NSUM 16

// scaled anchors: ANCHORS / (608/76)
__device__ __constant__ float c_aw[3] = {1.25f, 2.0f, 4.125f};
__device__ __constant__ float c_ah[3] = {1.625f, 3.75f, 2.875f};

typedef __attribute__((ext_vector_type(2))) float v2f;
typedef __attribute__((ext_vector_type(8))) float v8f;

// stable softplus: log(1+e^z)
__device__ __forceinline__ float sp(float z) {
    return fmaxf(z, 0.f) + log1pf(expf(-fabsf(z)));
}
// bce(sigmoid(z), t) with logs clipped at -100 (reference _bce)
__device__ __forceinline__ float bce_logit(float z, float t) {
    return t * fminf(sp(-z), 100.f) + (1.f - t) * fminf(sp(z), 100.f);
}

struct TgtInfo {
    bool valid, ok;
    int gi, gj, best, cls;
    float gx, gy, gw, gh;
    float iou[3];
    int cell;       // flat index into (B,A,H,W), best anchor
};

__device__ __forceinline__ TgtInfo decode_target(const float* tgt, int t) {
    TgtInfo r;
    int b = t / T_;
    const float* p = tgt + (size_t)t * 5;
    float cf = p[0], xx = p[1], yy = p[2], ww = p[3], hh = p[4];
    r.valid = (cf + xx + yy + ww + hh) != 0.f;
    r.cls = (int)cf;
    r.gx = xx * (float)W_;  r.gy = yy * (float)H_;
    r.gw = ww * (float)W_;  r.gh = hh * (float)H_;
    r.gi = (int)r.gx;       r.gj = (int)r.gy;
    r.ok = r.valid && (r.gj < H_) && (r.gi < W_);
    float a1 = (r.gw + 1.f) * (r.gh + 1.f);
    r.best = 0;
    float bestv = -1.f;
    #pragma unroll
    for (int a = 0; a < 3; a++) {
        float inter = fmaxf(fminf(r.gw, c_aw[a]) + 1.f, 0.f) *
                      fmaxf(fminf(r.gh, c_ah[a]) + 1.f, 0.f);
        float a2 = (c_aw[a] + 1.f) * (c_ah[a] + 1.f);
        r.iou[a] = inter / (a1 + a2 - inter + 1e-16f);
        if (r.iou[a] > bestv) { bestv = r.iou[a]; r.best = a; }
    }
    r.cell = (((b * A_ + r.best) * H_) + r.gj) * W_ + r.gi;
    return r;
}

// ---------------- kernel 1: init workspace ----------------
__global__ void k_init(float* mask, float* noobj, unsigned* clsbits, int* winner) {
    int i = blockIdx.x * blockDim.x + threadIdx.x;
    if (i < N_) { mask[i] = 0.f; noobj[i] = 1.f; winner[i] = -1; }
    if (i < 3 * N_) clsbits[i] = 0u;
}

// ---------------- kernel 2: build targets, pass 1 ----------------
__global__ void k_build1(const float* __restrict__ tgt, float* mask, float* noobj,
                         unsigned* clsbits, int* winner) {
    int t = blockIdx.x * blockDim.x + threadIdx.x;
    if (t >= NT_) return;
    TgtInfo r = decode_target(tgt, t);
    int b = t / T_;
    if (r.valid) {
        #pragma unroll
        for (int a = 0; a < 3; a++) {
            if (r.iou[a] > 0.5f) {   // IGNORE_THR, strict >
                int idx = (((b * A_ + a) * H_) + r.gj) * W_ + r.gi;
                if (idx >= 0 && idx < N_) noobj[idx] = 0.f;  // idempotent: deterministic
            }
        }
    }
    if (r.ok && r.cell >= 0 && r.cell < N_) {
        mask[r.cell] = 1.f;                         // idempotent
        atomicMax(&winner[r.cell], t);              // deterministic duplicate resolution
        if (r.cls >= 0 && r.cls < C_)
            atomicOr(&clsbits[r.cell * 3 + (r.cls >> 5)], 1u << (r.cls & 31));
    }
}

// ---------------- kernel 3: build targets, pass 2 (winner writes) ----------------
__global__ void k_build2(const float* __restrict__ tgt, const int* __restrict__ winner,
                         float* tx, float* ty, float* tw, float* th) {
    int t = blockIdx.x * blockDim.x + threadIdx.x;
    if (t >= NT_) return;
    TgtInfo r = decode_target(tgt, t);
    if (r.ok && r.cell >= 0 && r.cell < N_ && winner[r.cell] == t) {
        tx[r.cell] = r.gx - (float)r.gi;
        ty[r.cell] = r.gy - (float)r.gj;
        tw[r.cell] = logf(r.gw / c_aw[r.best] + 1e-16f);
        th[r.cell] = logf(r.gh / c_ah[r.best] + 1e-16f);
    }
}

// ---------------- kernel 4: dense reduction over B*A*H*W cells ----------------
__global__ __launch_bounds__(256)
void k_reduce(const float* __restrict__ in, const float* __restrict__ mask,
              const float* __restrict__ noobj, const float* __restrict__ tx,
              const float* __restrict__ ty, const float* __restrict__ tw,
              const float* __restrict__ th, const unsigned* __restrict__ clsbits,
              float* __restrict__ partials) {
    // s[0]=n_m s[1]=n_nm s[2..5]=x,y,w,h s[6]=conf_obj s[7]=conf_noobj s[8]=cls
    float s[9];
    #pragma unroll
    for (int k = 0; k < 9; k++) s[k] = 0.f;

    for (int n = blockIdx.x * blockDim.x + threadIdx.x; n < N_;
         n += gridDim.x * blockDim.x) {
        int ba  = n / HW_;            // b*A + a
        int rem = n - ba * HW_;       // j*W + i
        const float* base = in + (size_t)ba * 85 * HW_ + rem;
        float zc = base[4 * HW_];     // conf logit — the only dense channel
        float no = noobj[n];
        s[1] += no;
        s[7] += no * fminf(sp(zc), 100.f);   // -clip(log(1-sigmoid(zc)),-100)
        float m = mask[n];
        if (m != 0.f) {               // <= ~800 cells total
            s[0] += 1.f;
            float zx = base[0], zy = base[HW_];
            float zw = base[2 * HW_], zh = base[3 * HW_];
            s[2] += bce_logit(zx, tx[n]);
            s[3] += bce_logit(zy, ty[n]);
            float dw = zw - tw[n];  s[4] += dw * dw;
            float dh = zh - th[n];  s[5] += dh * dh;
            s[6] += fminf(sp(-zc), 100.f);   // bce(conf, 1)
            unsigned b0 = clsbits[n * 3 + 0];
            unsigned b1 = clsbits[n * 3 + 1];
            unsigned b2 = clsbits[n * 3 + 2];
            float sc = 0.f;
            for (int c = 0; c < C_; c++) {
                float z = base[(5 + c) * HW_];
                unsigned w32 = (c < 32) ? b0 : (c < 64) ? b1 : b2;
                float tt = (float)((w32 >> (c & 31)) & 1u);
                sc += bce_logit(z, tt);
            }
            s[8] += sc;
        }
    }

    __shared__ float sd[256 * 9];
    #pragma unroll
    for (int k = 0; k < 9; k++) sd[threadIdx.x * 9 + k] = s[k];
    __syncthreads();
    for (int str = 128; str > 0; str >>= 1) {
        if ((int)threadIdx.x < str) {
            #pragma unroll
            for (int k = 0; k < 9; k++)
                sd[threadIdx.x * 9 + k] += sd[(threadIdx.x + str) * 9 + k];
        }
        __syncthreads();
    }
    if (threadIdx.x == 0) {
        #pragma unroll
        for (int k = 0; k < 9; k++)  partials[blockIdx.x * NSUM + k] = sd[k];
        #pragma unroll
        for (int k = 9; k < NSUM; k++) partials[blockIdx.x * NSUM + k] = 0.f;
    }
}

// ---------------- kernel 5: WMMA column-sum of partials[64][16] + final losses ----
__global__ void k_final(const float* __restrict__ partials, float* __restrict__ out) {
    int lane = threadIdx.x;      // one wave32, EXEC all ones (WMMA requirement)
    int col  = lane & 15;
    int half = lane >> 4;
    v8f acc = {};
    v2f av;  av.x = 1.f;  av.y = 1.f;   // A = ones(16x4)
    #pragma unroll
    for (int chunk = 0; chunk < NBLK / 4; chunk++) {
        v2f bv;   // B = 4x16 slice of partials; any K permutation sums out under A=ones
        bv.x = partials[(chunk * 4 + half * 2 + 0) * NSUM + col];
        bv.y = partials[(chunk * 4 + half * 2 + 1) * NSUM + col];
        // D = ones(16x4) x B(4x16) + C  -> every D element in lane l = colsum[l&15]
        acc = __builtin_amdgcn_wmma_f32_16x16x4_f32(
                  false, av, false, bv, (short)0, acc, false, false);
    }
    __shared__ float sums[16];
    if (lane < 16) sums[lane] = acc[0];
    __syncthreads();
    if (lane == 0) {
        const float Nf = (float)N_;
        float n_m  = sums[0];
        float n_nm = sums[1];
        float lx = sums[2] / Nf / n_m;
        float ly = sums[3] / Nf / n_m;
        float lw = sums[4] / Nf / n_m;
        float lh = sums[5] / Nf / n_m;
        float lconf = sums[6] / Nf / n_m + 0.5f * (sums[7] / Nf / n_nm);
        float lcls  = sums[8] / (n_m * (float)C_) / n_m;
        float loss = 2.5f * (lx + ly) + 2.5f * (lw + lh) + lconf + lcls;
        out[0] = loss; out[1] = lx; out[2] = ly; out[3] = lw;
        out[4] = lh;   out[5] = lconf; out[6] = lcls;
    }
}

extern "C" void kernel_launch(void* const* d_in, const int* in_sizes, int n_in,
                              void* d_out, int out_size, void* d_ws, size_t ws_size,
                              hipStream_t stream) {
    const float* in  = (const float*)d_in[0];   // (B, A*85, H, W) f32
    const float* tgt = (const float*)d_in[1];   // (B, T, 5) f32
    float* out = (float*)d_out;                 // 7 f32 scalars

    // workspace layout (all 4-byte aligned): ~11.1 MB total
    float* ws       = (float*)d_ws;
    float* mask     = ws;                 // N
    float* noobj    = mask  + N_;         // N
    float* tx       = noobj + N_;         // N
    float* ty       = tx    + N_;         // N
    float* tw       = ty    + N_;         // N
    float* th       = tw    + N_;         // N
    float* partials = th    + N_;         // NBLK*NSUM
    unsigned* clsbits = (unsigned*)(partials + NBLK * NSUM);  // 3N
    int* winner     = (int*)(clsbits + 3 * N_);               // N

    k_init  <<<(3 * N_ + 255) / 256, 256, 0, stream>>>(mask, noobj, clsbits, winner);
    k_build1<<<(NT_ + 255) / 256,    256, 0, stream>>>(tgt, mask, noobj, clsbits, winner);
    k_build2<<<(NT_ + 255) / 256,    256, 0, stream>>>(tgt, winner, tx, ty, tw, th);
    k_reduce<<<NBLK,                 256, 0, stream>>>(in, mask, noobj, tx, ty, tw, th,
                                                       clsbits, partials);
    k_final <<<1,                     32, 0, stream>>>(partials, out);
}